// MoEGate_53721450938816
// MI455X (gfx1250) — compile-verified
//
#include <hip/hip_runtime.h>

typedef __attribute__((ext_vector_type(2))) float v2f;
typedef __attribute__((ext_vector_type(8))) float v8f;

#define TOKENS  32768
#define DIM     4096
#define EXPERTS 16
#define TILES   (TOKENS / 16)          // 2048
#define SPLITS  4                      // K split 4-way per tile
#define KSPAN   (DIM / SPLITS)         // 1024
#define TILES_PER_BLOCK 2              // 8 waves = 2 tiles x 4 splits

__device__ __forceinline__ bool topgt(float a, int ai, float b, int bi) {
    // "a beats b": larger value wins; ties -> smaller index first (jax top_k order)
    return (a > b) || (a == b && ai < bi);
}

__global__ __launch_bounds__(256) void moe_gate_wmma_f32(
    const float* __restrict__ x,        // (32768, 4096)
    const float* __restrict__ w,        // (16, 4096)
    float*       __restrict__ out_vals, // (32768, 2)
    int*         __restrict__ out_idx)  // (32768, 2)
{
    __shared__ float red[8][32][8];     // [wave][lane][acc] partial logits, 8 KB

    const int lane  = threadIdx.x & 31;
    const int wave  = threadIdx.x >> 5;
    const int split = wave & (SPLITS - 1);
    const int tile  = blockIdx.x * TILES_PER_BLOCK + (wave >> 2);

    const int hi = lane >> 4;   // 0: lanes 0-15, 1: lanes 16-31
    const int lo = lane & 15;

    const int kbase = split * KSPAN;

    // A operand (16x4 f32): lane holds row = lo, K-pair {k+2*hi, k+2*hi+1}
    const float* arow = x + (size_t)(tile * 16 + lo) * DIM + kbase + 2 * hi;
    // B operand (4x16 f32): B[k][n] = w[n][k]; lane holds col n = lo, same K-pair
    const float* brow = w + (size_t)lo * DIM + kbase + 2 * hi;

    v8f c = {};
    for (int kk = 0; kk < KSPAN; kk += 32) {
        v2f a[8], b[8];
#pragma unroll
        for (int u = 0; u < 8; ++u) {
            // activations: streamed once -> non-temporal; weights: cache-resident
            a[u] = __builtin_nontemporal_load((const v2f*)(arow + kk + 4 * u));
            b[u] = *(const v2f*)(brow + kk + 4 * u);
        }
#pragma unroll
        for (int u = 0; u < 8; ++u) {
            // (neg_a, A, neg_b, B, c_mod, C, reuse_a, reuse_b)
            c = __builtin_amdgcn_wmma_f32_16x16x4_f32(
                    false, a[u], false, b[u], (short)0, c, false, false);
        }
    }

    // Cross-wave split-K reduction through LDS.
#pragma unroll
    for (int j = 0; j < 8; ++j) red[wave][lane][j] = c[j];
    __syncthreads();
    if (split != 0) return;

#pragma unroll
    for (int s = 1; s < SPLITS; ++s)
#pragma unroll
        for (int j = 0; j < 8; ++j) c[j] += red[wave + s][lane][j];

    // c[j] = logit for token row (tile*16 + j + 8*hi), expert e = lo.
    // Softmax + top-2 across the 16 experts = across the 16-lane half-wave.
#pragma unroll
    for (int j = 0; j < 8; ++j) {
        const float v = c[j];
        float v1 = v, v2 = -__builtin_inff();
        int   i1 = lo, i2 = EXPERTS;

#pragma unroll
        for (int m = 1; m <= 8; m <<= 1) {
            float o1 = __shfl_xor(v1, m, 32);
            float o2 = __shfl_xor(v2, m, 32);
            int  oi1 = __shfl_xor(i1, m, 32);
            int  oi2 = __shfl_xor(i2, m, 32);
            float r1, r2; int ri1, ri2;
            if (topgt(v1, i1, o1, oi1)) {
                r1 = v1; ri1 = i1;
                if (topgt(o1, oi1, v2, i2)) { r2 = o1; ri2 = oi1; }
                else                        { r2 = v2; ri2 = i2;  }
            } else {
                r1 = o1; ri1 = oi1;
                if (topgt(v1, i1, o2, oi2)) { r2 = v1; ri2 = i1;  }
                else                        { r2 = o2; ri2 = oi2; }
            }
            v1 = r1; i1 = ri1; v2 = r2; i2 = ri2;
        }

        // softmax denominator: sum over all 16 experts of exp(logit - max)
        float s = __expf(v - v1);
#pragma unroll
        for (int m = 1; m <= 8; m <<= 1) s += __shfl_xor(s, m, 32);

        if (lo == j) {
            const int orow = tile * 16 + j + 8 * hi;
            const float inv = 1.0f / s;
            out_vals[orow * 2 + 0] = inv;                    // exp(v1 - v1) / s
            out_vals[orow * 2 + 1] = __expf(v2 - v1) * inv;
            out_idx [orow * 2 + 0] = i1;
            out_idx [orow * 2 + 1] = i2;
        }
    }
}

extern "C" void kernel_launch(void* const* d_in, const int* in_sizes, int n_in,
                              void* d_out, int out_size, void* d_ws, size_t ws_size,
                              hipStream_t stream) {
    (void)in_sizes; (void)n_in; (void)out_size; (void)d_ws; (void)ws_size;
    const float* x = (const float*)d_in[0];   // hidden_states (4,8192,4096) f32
    const float* w = (const float*)d_in[1];   // weight (16,4096) f32

    float* vals = (float*)d_out;                        // (32768, 2) f32
    int*   idx  = (int*)(vals + (size_t)TOKENS * 2);    // (32768, 2) i32 (bit pattern)

    dim3 grid(TILES / TILES_PER_BLOCK);  // 1024 blocks
    dim3 block(256);                     // 8 waves = 2 tiles x 4 K-splits
    moe_gate_wmma_f32<<<grid, block, 0, stream>>>(x, w, vals, idx);
}